// HeteroRGCNLayer_82952998355814
// MI455X (gfx1250) — compile-verified
//
#include <hip/hip_runtime.h>

#define N_USER 100000
#define N_ITEM 100000
#define E_NUM  500000
#define F      128

typedef __attribute__((ext_vector_type(2))) float v2f;
typedef __attribute__((ext_vector_type(8))) float v8f;

// ---------------------------------------------------------------------------
// Y[row, n] = sum_k X[row,k] * W[k,n] + b[n]
// One block = 16 rows x 128 cols. 8 waves; wave w owns N-tile [16w, 16w+16).
// K loop in steps of 4 using V_WMMA_F32_16X16X4_F32 (true f32, matches ref).
// ---------------------------------------------------------------------------
__global__ __launch_bounds__(256) void gemm_bias_wmma(
    const float* __restrict__ X, const float* __restrict__ W,
    const float* __restrict__ b, float* __restrict__ Y)
{
    __shared__ float sW[F * F];                    // 64 KB of 320 KB/WGP

    const int tid = threadIdx.x;
    // stage W into LDS (16384 floats = 4096 float4, 16 per thread)
    const float4* W4  = (const float4*)W;
    float4*       sW4 = (float4*)sW;
#pragma unroll
    for (int i = 0; i < (F * F / 4) / 256; ++i)
        sW4[tid + i * 256] = W4[tid + i * 256];
    __syncthreads();

    const int wave = tid >> 5;
    const int lane = tid & 31;
    const int nsel = lane & 15;       // N within tile (and M-row index for A)
    const int hi   = lane >> 4;       // lane group: selects K 0/1 vs K 2/3
    const int row0 = blockIdx.x * 16;
    const int n0   = wave * 16;

    // D = A*B + C : preload bias into accumulator (C depends only on N).
    v8f acc;
    const float bias = b[n0 + nsel];
#pragma unroll
    for (int r = 0; r < 8; ++r) acc[r] = bias;

    const float* Arow = X + (size_t)(row0 + nsel) * F;

#pragma unroll 4
    for (int k = 0; k < F; k += 4) {
        // A 16x4 f32: lanes0-15 -> {K=k,k+1}, lanes16-31 -> {K=k+2,k+3}, M=nsel
        v2f a = *(const v2f*)(Arow + k + hi * 2);
        // B 4x16 f32: vgpr0 row K=k(+2hi), vgpr1 row K=k+1(+2hi), col n0+nsel
        v2f bf;
        bf.x = sW[(k + hi * 2)     * F + n0 + nsel];
        bf.y = sW[(k + hi * 2 + 1) * F + n0 + nsel];
        acc = __builtin_amdgcn_wmma_f32_16x16x4_f32(
                  false, a, false, bf, (short)0, acc, false, false);
    }

    // C/D 16x16 f32 layout: VGPR r -> M = r + 8*hi, N = nsel
    float* Yp = Y + (size_t)row0 * F + n0;
#pragma unroll
    for (int r = 0; r < 8; ++r)
        Yp[(size_t)(r + hi * 8) * F + nsel] = acc[r];
}

// ---------------------------------------------------------------------------
__device__ __forceinline__ void atomic_add_f32(float* p, float v) {
    __hip_atomic_fetch_add(p, v, __ATOMIC_RELAXED, __HIP_MEMORY_SCOPE_AGENT);
}

// one thread per edge: deg[dst] += 1
__global__ __launch_bounds__(256) void count_deg(
    const int* __restrict__ dst, float* __restrict__ cnt, int n)
{
    int e = blockIdx.x * blockDim.x + threadIdx.x;
    if (e < n) atomic_add_f32(&cnt[dst[e]], 1.0f);
}

// 128 threads per edge: out[dst, c] += Wh[src, c] / max(deg[dst], 1)
__global__ __launch_bounds__(256) void scatter_mean(
    const float* __restrict__ Wh, const int* __restrict__ src,
    const int* __restrict__ dst, const float* __restrict__ cnt,
    float* __restrict__ out)
{
    int t = blockIdx.x * blockDim.x + threadIdx.x;
    int e = t >> 7;          // edge index (grid sized exactly: E*128/256 blocks)
    int c = t & (F - 1);     // feature index
    int s = src[e];
    int d = dst[e];
    float inv = 1.0f / fmaxf(cnt[d], 1.0f);
    float v   = Wh[(size_t)s * F + c] * inv;
    atomic_add_f32(&out[(size_t)d * F + c], v);
}

// ---------------------------------------------------------------------------
extern "C" void kernel_launch(void* const* d_in, const int* in_sizes, int n_in,
                              void* d_out, int out_size, void* d_ws, size_t ws_size,
                              hipStream_t stream)
{
    const float* feat_user = (const float*)d_in[0];
    const float* feat_item = (const float*)d_in[1];
    const int*   src0 = (const int*)d_in[2];
    const int*   dst0 = (const int*)d_in[3];
    const int*   src1 = (const int*)d_in[4];
    const int*   dst1 = (const int*)d_in[5];
    const int*   src2 = (const int*)d_in[6];
    const int*   dst2 = (const int*)d_in[7];
    const float* W0 = (const float*)d_in[8];
    const float* b0 = (const float*)d_in[9];
    const float* W1 = (const float*)d_in[10];
    const float* b1 = (const float*)d_in[11];
    const float* W2 = (const float*)d_in[12];
    const float* b2 = (const float*)d_in[13];

    float* ws   = (float*)d_ws;
    float* Wh0  = ws;                                // [N_USER, F]
    float* Wh1  = Wh0  + (size_t)N_USER * F;         // [N_USER, F]
    float* Wh2  = Wh1  + (size_t)N_USER * F;         // [N_ITEM, F]
    float* cnt0 = Wh2  + (size_t)N_ITEM * F;         // [N_USER]
    float* cnt1 = cnt0 + N_USER;                     // [N_ITEM]
    float* cnt2 = cnt1 + N_ITEM;                     // [N_USER]

    float* out_user = (float*)d_out;                 // [N_USER, F]
    float* out_item = out_user + (size_t)N_USER * F; // [N_ITEM, F]

    // zero accumulators (capturable memset nodes)
    hipMemsetAsync(d_out, 0, (size_t)out_size * sizeof(float), stream);
    hipMemsetAsync(cnt0, 0, (size_t)(2 * N_USER + N_ITEM) * sizeof(float), stream);

    dim3 blk(256);

    // per-etype projections (f32 WMMA)
    gemm_bias_wmma<<<N_USER / 16, blk, 0, stream>>>(feat_user, W0, b0, Wh0);
    gemm_bias_wmma<<<N_USER / 16, blk, 0, stream>>>(feat_user, W1, b1, Wh1);
    gemm_bias_wmma<<<N_ITEM / 16, blk, 0, stream>>>(feat_item, W2, b2, Wh2);

    // degree counts
    int cblocks = (E_NUM + 255) / 256;
    count_deg<<<cblocks, blk, 0, stream>>>(dst0, cnt0, E_NUM);
    count_deg<<<cblocks, blk, 0, stream>>>(dst1, cnt1, E_NUM);
    count_deg<<<cblocks, blk, 0, stream>>>(dst2, cnt2, E_NUM);

    // mean-aggregate with divide folded into the scatter; user dst relations
    // (etype0 + etype2) both accumulate into out_user -> cross-relation sum.
    int sblocks = (E_NUM * F) / 256;                 // exact: 250000
    scatter_mean<<<sblocks, blk, 0, stream>>>(Wh0, src0, dst0, cnt0, out_user);
    scatter_mean<<<sblocks, blk, 0, stream>>>(Wh2, src2, dst2, cnt2, out_user);
    scatter_mean<<<sblocks, blk, 0, stream>>>(Wh1, src1, dst1, cnt1, out_item);
}